// RBFUnit_73409581023396
// MI455X (gfx1250) — compile-verified
//
#include <hip/hip_runtime.h>

typedef __attribute__((ext_vector_type(2))) float v2f;
typedef __attribute__((ext_vector_type(8))) float v8f;

#define DIM     84
#define KPROTO  10
#define CHUNKS  21          // 84 / 4 K-slices per 16x16 tile

// ds_swizzle group-of-32: xor=0x10, or=0, and=0x1f  -> swap lanes L <-> L+16
#define SWAPX16 0x401F

__global__ __launch_bounds__(256) void rbf_dist2_wmma(
    const float* __restrict__ x,      // [N, 84]
    const float* __restrict__ w,      // [10, 84]
    float* __restrict__ out,          // [N, 10]
    int n_rows)
{
    const int lane  = threadIdx.x & 31;
    const int wave  = blockIdx.x * (blockDim.x >> 5) + (threadIdx.x >> 5);
    const int row0  = wave * 16;                 // wave-uniform tile guard
    if (row0 >= n_rows) return;

    const int nIdx = lane & 15;                  // output column / row-in-tile
    const int hi   = lane >> 4;                  // 0: lanes 0-15, 1: lanes 16-31
    const int koff = hi * 2;                     // K phase per lane half

    // ---- B matrix: (-2 * weight) in WMMA B layout, register-resident -------
    // VGPR0: lanes 0-15 K=4c+0, lanes 16-31 K=4c+2 ; VGPR1: K=4c+1 / K=4c+3
    v2f  b[CHUNKS];
    float w2p = 0.0f;
    const bool valid_col = (nIdx < KPROTO);
    #pragma unroll
    for (int c = 0; c < CHUNKS; ++c) {
        v2f bw = {0.0f, 0.0f};
        if (valid_col) {
            const float* p = w + nIdx * DIM + 4 * c + koff;
            bw.x = -2.0f * p[0];
            bw.y = -2.0f * p[1];
        }
        b[c] = bw;
        w2p += bw.x * bw.x + bw.y * bw.y;        // = 4 * partial(w^2)
    }
    // combine the two K-half partials (lanes L and L+16 both hold column nIdx)
    w2p += __int_as_float(__builtin_amdgcn_ds_swizzle(__float_as_int(w2p), SWAPX16));
    const float w2 = 0.25f * w2p;                // (-2w)^2 = 4 w^2

    // ---- C init: C[m][n] = w2[n] for all m ---------------------------------
    v8f acc;
    #pragma unroll
    for (int r = 0; r < 8; ++r) acc[r] = w2;

    // ---- stream A (x tile) and accumulate  -2 * x.w  -----------------------
    const float* rowp = x + (size_t)(row0 + nIdx) * DIM + koff;
    float x2p = 0.0f;
    #pragma unroll
    for (int c = 0; c < CHUNKS; ++c) {
        v2f a;
        a.x = rowp[4 * c + 0];
        a.y = rowp[4 * c + 1];
        x2p += a.x * a.x + a.y * a.y;            // free x^2 partial
        acc = __builtin_amdgcn_wmma_f32_16x16x4_f32(
                  false, a, false, b[c], (short)0, acc, false, false);
    }
    // total x^2 per row (combine K-half partials across lane 16 swap)
    x2p += __int_as_float(__builtin_amdgcn_ds_swizzle(__float_as_int(x2p), SWAPX16));

    // ---- augmented K slice: A[:,K=84] = x2[m], B[K=84,:] = 1  --------------
    // adds x2[m] to every column: acc[m][n] = x2[m] + w2[n] - 2 x.w
    v2f a_ext; a_ext.x = hi ? 0.0f : x2p;  a_ext.y = 0.0f;
    v2f b_ext; b_ext.x = hi ? 0.0f : 1.0f; b_ext.y = 0.0f;
    acc = __builtin_amdgcn_wmma_f32_16x16x4_f32(
              false, a_ext, false, b_ext, (short)0, acc, false, false);

    // ---- store: lane -> column nIdx, rows m = r + 8*hi (C/D layout) --------
    if (valid_col) {
        #pragma unroll
        for (int r = 0; r < 8; ++r) {
            const int m = r + 8 * hi;
            out[(size_t)(row0 + m) * KPROTO + nIdx] = acc[r];
        }
    }
}

extern "C" void kernel_launch(void* const* d_in, const int* in_sizes, int n_in,
                              void* d_out, int out_size, void* d_ws, size_t ws_size,
                              hipStream_t stream) {
    const float* x = (const float*)d_in[0];   // [524288, 84]
    const float* w = (const float*)d_in[1];   // [10, 84]
    float* out     = (float*)d_out;           // [524288, 10]

    const int n_rows = in_sizes[0] / DIM;
    const int waves  = (n_rows + 15) / 16;
    const int wpb    = 256 / 32;              // 8 waves per block
    const int blocks = (waves + wpb - 1) / wpb;

    rbf_dist2_wmma<<<blocks, 256, 0, stream>>>(x, w, out, n_rows);
}